// TaskAttentionPool_89910845375281
// MI455X (gfx1250) — compile-verified
//
#include <hip/hip_runtime.h>
#include <hip/hip_bf16.h>

// Problem constants (match reference)
#define DIM   512
#define NH    8
#define NT    4
#define HD    64
#define NTOK  8192
#define NB    16
#define KSEL  64
#define TH32  32   // NT*NH combined query columns

typedef float v2f __attribute__((ext_vector_type(2)));
typedef float v8f __attribute__((ext_vector_type(8)));

// ---------------------------------------------------------------------------
// Pass A: query-side setup (tiny). Computes, for th = t*NH + h:
//   qk[d][th]  = sum_hd qp[t, h*64+hd] * wk[h*64+hd, d]          (then *1/8)
//   qkg[th][d] = qk*g_d            (row-major [32][512], for WMMA B frags)
//   sqg[th]    = sum_d qk*g_d
//   cb[th]     = sum_d qk*b_d + qbias[th]/8
// where qp = LN(q)/tau @ wq^T + bq.
// ---------------------------------------------------------------------------
__global__ __launch_bounds__(512) void pass_a(
    const float* __restrict__ q, const float* __restrict__ w_in,
    const float* __restrict__ b_in, const float* __restrict__ pre_g,
    const float* __restrict__ pre_b, const float* __restrict__ q_g,
    const float* __restrict__ q_b, const float* __restrict__ log_tau,
    float* __restrict__ qkg, float* __restrict__ cb, float* __restrict__ sqg)
{
    __shared__ float red[512];
    __shared__ float qln[NT][DIM];
    __shared__ float qp[NT][DIM];
    __shared__ float qbias[TH32];
    const int tid = threadIdx.x;

    float tau = __expf(log_tau[0]);
    tau = fminf(fmaxf(tau, 0.1f), 10.0f);
    const float inv_tau = 1.0f / tau;

    // LayerNorm of the NT query rows
    for (int t = 0; t < NT; ++t) {
        float x = q[t * DIM + tid];
        red[tid] = x; __syncthreads();
        for (int s = 256; s > 0; s >>= 1) { if (tid < s) red[tid] += red[tid + s]; __syncthreads(); }
        float mu = red[0] * (1.0f / DIM); __syncthreads();
        float dx = x - mu;
        red[tid] = dx * dx; __syncthreads();
        for (int s = 256; s > 0; s >>= 1) { if (tid < s) red[tid] += red[tid + s]; __syncthreads(); }
        float var = red[0] * (1.0f / DIM); __syncthreads();
        float r = rsqrtf(var + 1e-5f);
        qln[t][tid] = (dx * r * q_g[tid] + q_b[tid]) * inv_tau;
    }
    __syncthreads();

    // qp[t][i] = qln[t] . wq_row_i + bq[i]
    for (int t = 0; t < NT; ++t) {
        float acc = b_in[tid];
        const float* wrow = w_in + (size_t)tid * DIM;
        for (int j = 0; j < DIM; ++j) acc += qln[t][j] * wrow[j];
        qp[t][tid] = acc;
    }
    __syncthreads();

    // qbias[th] = qp[t, h*64:...] . bk_h
    if (tid < TH32) {
        int t = tid / NH, h = tid % NH;
        float acc = 0.f;
        for (int hd = 0; hd < HD; ++hd)
            acc += qp[t][h * HD + hd] * b_in[DIM + h * HD + hd];
        qbias[tid] = acc;
    }
    __syncthreads();

    // qk columns + folded constants (per th: 64-MAC dot, then 2 block reductions)
    for (int th = 0; th < TH32; ++th) {
        int t = th / NH, h = th % NH;
        float acc = 0.f;
        for (int hd = 0; hd < HD; ++hd)
            acc += qp[t][h * HD + hd] * w_in[(size_t)(DIM + h * HD + hd) * DIM + tid];
        float qks = acc * 0.125f;            // 1/sqrt(HD)
        qkg[th * DIM + tid] = qks * pre_g[tid];

        red[tid] = qks * pre_b[tid]; __syncthreads();
        for (int s = 256; s > 0; s >>= 1) { if (tid < s) red[tid] += red[tid + s]; __syncthreads(); }
        if (tid == 0) cb[th] = red[0] + qbias[th] * 0.125f;
        __syncthreads();

        red[tid] = qks * pre_g[tid]; __syncthreads();
        for (int s = 256; s > 0; s >>= 1) { if (tid < s) red[tid] += red[tid + s]; __syncthreads(); }
        if (tid == 0) sqg[th] = red[0];
        __syncthreads();
    }
}

// ---------------------------------------------------------------------------
// Pass B: fused LayerNorm-stats + skinny WMMA GEMM over raw tokens.
// One wave = 16 tokens x 32 score columns, K=512 via v_wmma_f32_16x16x4_f32.
// A fragment (f32 16x4): lane L -> M = L%16, K = k0 + r + 2*(L/16).
// B fragment (f32 4x16): lane L -> N = L%16, same K mapping (qkg row-major).
// Sum(x), Sum(x^2) accumulated from the same A loads; halves merged with
// shfl_xor(16); final score = rstd*(raw - mu*sqg) + cb.
// ---------------------------------------------------------------------------
__global__ __launch_bounds__(256) void pass_b(
    const float* __restrict__ tokens, const float* __restrict__ qkg,
    const float* __restrict__ cb, const float* __restrict__ sqg,
    float* __restrict__ sc)
{
    const int b    = blockIdx.x;
    const int wave = threadIdx.x >> 5;
    const int lane = threadIdx.x & 31;
    const int g    = lane >> 4;
    const int m    = lane & 15;
    const int tokenBase = blockIdx.y * 128 + wave * 16;

    const float* arow  = tokens + ((size_t)b * NTOK + tokenBase + m) * DIM + 2 * g;
    const float* brow0 = qkg + (size_t)m * DIM + 2 * g;
    const float* brow1 = qkg + (size_t)(m + 16) * DIM + 2 * g;

    v8f c0 = {}; v8f c1 = {};
    float sx = 0.f, sxx = 0.f;
    for (int k0 = 0; k0 < DIM; k0 += 4) {
        v2f a  = __builtin_nontemporal_load((const v2f*)(arow + k0));
        v2f b0 = *(const v2f*)(brow0 + k0);
        v2f b1 = *(const v2f*)(brow1 + k0);
        sx  += a.x + a.y;
        sxx += a.x * a.x + a.y * a.y;
        c0 = __builtin_amdgcn_wmma_f32_16x16x4_f32(false, a, false, b0, (short)0, c0, false, false);
        c1 = __builtin_amdgcn_wmma_f32_16x16x4_f32(false, a, false, b1, (short)0, c1, false, false);
    }
    // lanes L and L^16 each saw half of token (L%16)'s row
    sx  += __shfl_xor(sx, 16);
    sxx += __shfl_xor(sxx, 16);
    float mu   = sx * (1.0f / DIM);
    float var  = sxx * (1.0f / DIM) - mu * mu;
    float rstd = rsqrtf(fmaxf(var, 0.f) + 1e-5f);

    const float cb0 = cb[m],  cb1 = cb[m + 16];
    const float sq0 = sqg[m], sq1 = sqg[m + 16];

    #pragma unroll
    for (int v = 0; v < 8; ++v) {
        int tok   = v + 8 * g;                 // C row within the 16-token tile
        float muv = __shfl(mu, tok);
        float rsv = __shfl(rstd, tok);
        int n = tokenBase + tok;
        sc[((size_t)b * TH32 + m) * NTOK + n]      = rsv * (c0[v] - muv * sq0) + cb0;
        sc[((size_t)b * TH32 + m + 16) * NTOK + n] = rsv * (c1[v] - muv * sq1) + cb1;
    }
}

// ---------------------------------------------------------------------------
// Pass C: per (b,t): softmax over n per head, mean over heads, mask,
// normalize, prune (<1e-3), fallback one-hot, top-64 extraction, then
// m[b*NT+t] = sum_k w_k * LN(tokens[b, sel_k]).
// ---------------------------------------------------------------------------
__global__ __launch_bounds__(256) void pass_c(
    const float* __restrict__ sc, const unsigned char* __restrict__ mask,
    const float* __restrict__ tokens, const float* __restrict__ pre_g,
    const float* __restrict__ pre_b, float* __restrict__ mvec_out)
{
    __shared__ float alpha[NTOK];
    __shared__ float red[256];
    __shared__ int   redi[256];
    __shared__ float selVal[KSEL];
    __shared__ int   selIdx[KSEL];
    __shared__ float bcast;

    const int tid = threadIdx.x;
    const int b = blockIdx.x >> 2;
    const int t = blockIdx.x & 3;
    const unsigned char* mk = mask + (size_t)b * NTOK;

    for (int n = tid; n < NTOK; n += 256) alpha[n] = 0.f;
    __syncthreads();

    for (int h = 0; h < NH; ++h) {
        const float* s = sc + ((size_t)b * TH32 + t * NH + h) * NTOK;
        float mx = -3.0e38f;
        for (int n = tid; n < NTOK; n += 256) if (!mk[n]) mx = fmaxf(mx, s[n]);
        red[tid] = mx; __syncthreads();
        for (int st = 128; st > 0; st >>= 1) { if (tid < st) red[tid] = fmaxf(red[tid], red[tid + st]); __syncthreads(); }
        float smax = red[0]; __syncthreads();

        float ls = 0.f;
        for (int n = tid; n < NTOK; n += 256) if (!mk[n]) ls += __expf(s[n] - smax);
        red[tid] = ls; __syncthreads();
        for (int st = 128; st > 0; st >>= 1) { if (tid < st) red[tid] += red[tid + st]; __syncthreads(); }
        float inv = 1.0f / (8.0f * red[0]); __syncthreads();

        for (int n = tid; n < NTOK; n += 256) if (!mk[n]) alpha[n] += __expf(s[n] - smax) * inv;
        __syncthreads();
    }

    // alpha /= max(sum, EPS)
    float lsum = 0.f;
    for (int n = tid; n < NTOK; n += 256) lsum += alpha[n];
    red[tid] = lsum; __syncthreads();
    for (int st = 128; st > 0; st >>= 1) { if (tid < st) red[tid] += red[tid + st]; __syncthreads(); }
    float inv1 = 1.0f / fmaxf(red[0], 1e-8f); __syncthreads();
    for (int n = tid; n < NTOK; n += 256) alpha[n] *= inv1;
    __syncthreads();

    // pruned sum
    float ps = 0.f;
    for (int n = tid; n < NTOK; n += 256) { float a = alpha[n]; if (!mk[n] && a >= 0.001f) ps += a; }
    red[tid] = ps; __syncthreads();
    for (int st = 128; st > 0; st >>= 1) { if (tid < st) red[tid] += red[tid + st]; __syncthreads(); }
    float s2 = red[0]; __syncthreads();

    if (s2 <= 1e-8f) {
        // fallback: one-hot at argmax of (mask ? -1 : alpha), first index on ties
        float bv = -2.f; int bi = NTOK;
        for (int n = tid; n < NTOK; n += 256) {
            float v = mk[n] ? -1.f : alpha[n];
            if (v > bv || (v == bv && n < bi)) { bv = v; bi = n; }
        }
        red[tid] = bv; redi[tid] = bi; __syncthreads();
        for (int st = 128; st > 0; st >>= 1) {
            if (tid < st) {
                if (red[tid + st] > red[tid] || (red[tid + st] == red[tid] && redi[tid + st] < redi[tid])) {
                    red[tid] = red[tid + st]; redi[tid] = redi[tid + st];
                }
            }
            __syncthreads();
        }
        int amax = redi[0]; __syncthreads();
        for (int n = tid; n < NTOK; n += 256) alpha[n] = (n == amax) ? 1.f : 0.f;
    } else {
        for (int n = tid; n < NTOK; n += 256) {
            float a = alpha[n];
            alpha[n] = (!mk[n] && a >= 0.001f) ? a : 0.f;
        }
    }
    __syncthreads();

    // iterative top-64 (value desc, index asc on ties — matches lax.top_k)
    for (int k = 0; k < KSEL; ++k) {
        float bv = -1.f; int bi = NTOK;
        for (int n = tid; n < NTOK; n += 256) {
            float v = alpha[n];
            if (v > bv || (v == bv && n < bi)) { bv = v; bi = n; }
        }
        red[tid] = bv; redi[tid] = bi; __syncthreads();
        for (int st = 128; st > 0; st >>= 1) {
            if (tid < st) {
                if (red[tid + st] > red[tid] || (red[tid + st] == red[tid] && redi[tid + st] < redi[tid])) {
                    red[tid] = red[tid + st]; redi[tid] = redi[tid + st];
                }
            }
            __syncthreads();
        }
        if (tid == 0) {
            selVal[k] = red[0]; selIdx[k] = redi[0];
            if (redi[0] < NTOK) alpha[redi[0]] = -1.f;
        }
        __syncthreads();
    }

    if (tid == 0) {
        float ws = 0.f;
        for (int k = 0; k < KSEL; ++k) ws += selVal[k];
        bcast = 1.0f / fmaxf(ws, 1e-8f);
    }
    __syncthreads();
    const float invw = bcast;

    // m = sum_k w_k * LN(tokens[b, sel_k])   (each thread owns dims tid, tid+256)
    float macc0 = 0.f, macc1 = 0.f;
    for (int k = 0; k < KSEL; ++k) {
        float w = selVal[k] * invw;
        if (w > 0.f) {                        // uniform: selVal is shared
            const float* x = tokens + ((size_t)b * NTOK + selIdx[k]) * DIM;
            float xa = x[tid], xb = x[tid + 256];
            red[tid] = xa + xb; __syncthreads();
            for (int st = 128; st > 0; st >>= 1) { if (tid < st) red[tid] += red[tid + st]; __syncthreads(); }
            float mu = red[0] * (1.0f / DIM); __syncthreads();
            float da = xa - mu, db = xb - mu;
            red[tid] = da * da + db * db; __syncthreads();
            for (int st = 128; st > 0; st >>= 1) { if (tid < st) red[tid] += red[tid + st]; __syncthreads(); }
            float rstd = rsqrtf(red[0] * (1.0f / DIM) + 1e-5f); __syncthreads();
            macc0 += w * (da * rstd * pre_g[tid] + pre_b[tid]);
            macc1 += w * (db * rstd * pre_g[tid + 256] + pre_b[tid + 256]);
        }
    }
    mvec_out[(size_t)blockIdx.x * DIM + tid]       = macc0;
    mvec_out[(size_t)blockIdx.x * DIM + tid + 256] = macc1;
}

// ---------------------------------------------------------------------------
// Pass D: pooled = m (64x512) @ wv^T (512x512) + bv, via f32 WMMA.
// One wave per 16x16 output tile; K loop of 128 x (16x16x4).
// ---------------------------------------------------------------------------
__global__ __launch_bounds__(32) void pass_d(
    const float* __restrict__ mvec, const float* __restrict__ w_in,
    const float* __restrict__ b_in, float* __restrict__ out)
{
    const int lane = threadIdx.x;
    const int g = lane >> 4, mcol = lane & 15;
    const int Mt = blockIdx.x, Nt = blockIdx.y;

    const float* arow = mvec + (size_t)(Mt * 16 + mcol) * DIM + 2 * g;
    const float* brow = w_in + (size_t)(2 * DIM + Nt * 16 + mcol) * DIM + 2 * g;  // wv rows

    v8f c = {};
    for (int k0 = 0; k0 < DIM; k0 += 4) {
        v2f a  = *(const v2f*)(arow + k0);
        v2f bb = *(const v2f*)(brow + k0);
        c = __builtin_amdgcn_wmma_f32_16x16x4_f32(false, a, false, bb, (short)0, c, false, false);
    }
    const float bias = b_in[2 * DIM + Nt * 16 + mcol];
    #pragma unroll
    for (int v = 0; v < 8; ++v) {
        int row = Mt * 16 + v + 8 * g;
        out[(size_t)row * DIM + Nt * 16 + mcol] = c[v] + bias;
    }
}

// ---------------------------------------------------------------------------
extern "C" void kernel_launch(void* const* d_in, const int* in_sizes, int n_in,
                              void* d_out, int out_size, void* d_ws, size_t ws_size,
                              hipStream_t stream) {
    (void)in_sizes; (void)n_in; (void)out_size; (void)ws_size;
    const float*         tokens  = (const float*)d_in[0];
    const unsigned char* mask    = (const unsigned char*)d_in[1];  // jnp bool -> 1 byte/elem
    const float*         q       = (const float*)d_in[2];
    const float*         w_in    = (const float*)d_in[3];
    const float*         b_in    = (const float*)d_in[4];
    const float*         pre_g   = (const float*)d_in[5];
    const float*         pre_b   = (const float*)d_in[6];
    const float*         q_g     = (const float*)d_in[7];
    const float*         q_b     = (const float*)d_in[8];
    const float*         log_tau = (const float*)d_in[9];
    float* out = (float*)d_out;

    // workspace layout
    float* sc   = (float*)d_ws;                        // [16][32][8192] scores
    float* qkg  = sc + (size_t)NB * TH32 * NTOK;       // [32][512]
    float* cbv  = qkg + TH32 * DIM;                    // [32]
    float* sqg  = cbv + TH32;                          // [32]
    float* mvec = sqg + TH32;                          // [64][512]

    pass_a<<<1, 512, 0, stream>>>(q, w_in, b_in, pre_g, pre_b, q_g, q_b, log_tau,
                                  qkg, cbv, sqg);
    pass_b<<<dim3(NB, NTOK / 128), 256, 0, stream>>>(tokens, qkg, cbv, sqg, sc);
    pass_c<<<NB * NT, 256, 0, stream>>>(sc, mask, tokens, pre_g, pre_b, mvec);
    pass_d<<<dim3(NB * NT / 16, DIM / 16), 32, 0, stream>>>(mvec, w_in, b_in, out);
}